// WavGCRN_43078521979278
// MI455X (gfx1250) — compile-verified
//
#include <hip/hip_runtime.h>
#include <math.h>

// ---------------------------------------------------------------------------
// WavGCRN for MI455X (gfx1250).
// ~520 GFLOP fp32; all operand panels L2-resident (192MB) -> compute bound.
// Heavy math via V_WMMA_F32_16X16X4_F32 (fp32 matrix path = reference
// precision). ~200KB LDS per workgroup (CDNA5: 320KB/WGP).
// Adjacency matrices are pre-normalized/scaled/transposed/zero-padded once
// per phase so the WMMA K-loops are completely branch-free.
// ---------------------------------------------------------------------------

#define N_NODES 207
#define NPAD    208
#define HID     64
#define CIN     66            // 2 + HID
#define KPADP   68            // proj K per segment, padded
#define LSTRIDE 82            // even stride: conflict-free + b64-aligned pairs
#define BUFSZ   (NPAD * LSTRIDE)
#define BATCH   128
#define ALPHA_C 0.05f
#define BETA_C  0.95f
#define GAMMA_C 0.95f

typedef float v2f __attribute__((ext_vector_type(2)));
typedef float v8f __attribute__((ext_vector_type(8)));

__device__ __forceinline__ float geluf(float x) {
    return 0.5f * x * (1.0f + erff(x * 0.70710678118654752440f));
}

// ---------------------------------------------------------------------------
// Row/column sum reciprocals of (md + I).
// ---------------------------------------------------------------------------
__global__ void sums_kernel(const float* __restrict__ md,
                            float* __restrict__ rsinv,
                            float* __restrict__ csinv) {
    int b = blockIdx.x;
    int v = threadIdx.x;
    if (v >= N_NODES) return;
    const float* M = md + (size_t)b * N_NODES * N_NODES;
    float s = 1.0f;
    if (blockIdx.y == 0) {
        for (int w = 0; w < N_NODES; ++w) s += M[v * N_NODES + w];
        rsinv[b * N_NODES + v] = 1.0f / s;
    } else {
        for (int w = 0; w < N_NODES; ++w) s += M[w * N_NODES + v];
        csinv[b * N_NODES + v] = 1.0f / s;
    }
}

// ---------------------------------------------------------------------------
// Apad[b][m][k]  = BETA * norm(md+I)[k][m]          (A-operand layout, padded)
// ATpad[b][m][k] = BETA * norm((md+I)^T)[k][m]
// grid (B, NPAD)
// ---------------------------------------------------------------------------
__global__ void prep_adj_kernel(const float* __restrict__ md,
                                const float* __restrict__ rsinv,
                                const float* __restrict__ csinv,
                                float* __restrict__ Apad,
                                float* __restrict__ ATpad) {
    int b = blockIdx.x;
    int m = blockIdx.y;
    float* ar  = Apad  + ((size_t)b * NPAD + m) * NPAD;
    float* atr = ATpad + ((size_t)b * NPAD + m) * NPAD;
    for (int k = threadIdx.x; k < NPAD; k += blockDim.x) {
        float a = 0.0f, at = 0.0f;
        if (m < N_NODES && k < N_NODES) {
            float dg = (k == m) ? 1.0f : 0.0f;
            a  = (md[((size_t)b * N_NODES + k) * N_NODES + m] + dg) *
                 rsinv[b * N_NODES + k] * BETA_C;
            at = (md[((size_t)b * N_NODES + m) * N_NODES + k] + dg) *
                 csinv[b * N_NODES + k] * BETA_C;
        }
        ar[k] = a;
        atr[k] = at;
    }
}

// pAT[m][k] = GAMMA * pA[k][m], zero padded to 208x208
__global__ void prep_pa_kernel(const float* __restrict__ pA,
                               float* __restrict__ pAT) {
    int idx = blockIdx.x * blockDim.x + threadIdx.x;
    if (idx >= NPAD * NPAD) return;
    int m = idx / NPAD, k = idx - m * NPAD;
    pAT[idx] = (m < N_NODES && k < N_NODES) ? pA[k * N_NODES + m] * GAMMA_C
                                            : 0.0f;
}

// Wpad[g][seg*68+k][n] = (k<66) ? gW[g][seg*66+k][n] : 0
__global__ void prep_w_kernel(const float* __restrict__ gW,
                              float* __restrict__ Wpad) {
    int idx = blockIdx.x * blockDim.x + threadIdx.x;
    if (idx >= 18 * 3 * KPADP * HID) return;
    int n = idx & (HID - 1);
    int r = idx >> 6;                 // g*204 + seg*68 + k
    int g = r / (3 * KPADP);
    int sk = r - g * (3 * KPADP);
    int seg = sk / KPADP, k = sk - seg * KPADP;
    Wpad[idx] = (k < CIN)
        ? gW[((size_t)g * (3 * CIN) + seg * CIN + k) * HID + n]
        : 0.0f;
}

__global__ void fill0_kernel(float* __restrict__ p, int n) {
    int i = blockIdx.x * blockDim.x + threadIdx.x;
    if (i < n) p[i] = 0.0f;
}

// ---------------------------------------------------------------------------
// Per-step 2-channel input x_t (see round-0 comments).
// ---------------------------------------------------------------------------
__global__ void build_xt_kernel(const float* __restrict__ inp,
                                const float* __restrict__ ycl,
                                const float* __restrict__ prev,
                                float* __restrict__ xt,
                                int t, int mode, float sgn) {
    int idx = blockIdx.x * blockDim.x + threadIdx.x;
    if (idx >= BATCH * N_NODES) return;
    int b = idx / N_NODES, v = idx - b * N_NODES;
    float x0, x1;
    if (mode == 0) {
        const float* f = inp + ((size_t)(b * 2 + 0) * N_NODES + v) * 12;
        const float* T = inp + ((size_t)(b * 2 + 1) * N_NODES + v) * 12;
        x0 = (f[2 * t] + sgn * f[2 * t + 1]) * 0.70710678118654752440f;
        x1 = T[2 * t];
    } else {
        x0 = (t == 0) ? 0.0f : prev[((size_t)b * 12 + (t - 1)) * N_NODES + v];
        x1 = ycl[((size_t)(b * 2 + 1) * N_NODES + v) * 12 + t];
    }
    xt[idx * 2 + 0] = x0;
    xt[idx * 2 + 1] = x1;
}

// ---------------------------------------------------------------------------
// One GCN-GRU gate:
//   out = act( gcn(comb,A,pA0,W[base]) + gcn(comb,A^T,pA1,W[base+1]) + b )
// comb = [x_t | (R?R*H:H)] staged in LDS (208x82, zero padded).
// Propagation: D += (Apad-row ⊗ ldsB) + (pAT-row ⊗ ldsB): both adjacency
// matrices share one LDS B-operand fetch per k-step; K-loops branch-free.
// WMMA operand layouts per ISA 7.12.2 (A: m=lane&15, k=k0+2*hi{,+1};
// B: k rows as A; D: row=i+8*hi, col=lane&15).
// ---------------------------------------------------------------------------
__global__ __launch_bounds__(256)
void gate_kernel(const float* __restrict__ xt,
                 const float* __restrict__ H,
                 const float* __restrict__ Rg,       // null -> comb=[x|H]
                 const float* __restrict__ Apad,     // (B,208,208) pre-scaled
                 const float* __restrict__ ATpad,    // (B,208,208)
                 const float* __restrict__ pA0T,     // (208,208)
                 const float* __restrict__ pA1T,
                 const float* __restrict__ Wpad,     // (18,204,64)
                 const float* __restrict__ gB,       // (18,64)
                 int base0, int act,                 // act: 0 sigmoid, 1 tanh
                 float* __restrict__ out0,
                 float* __restrict__ out1) {
    extern __shared__ float lds[];                  // 3 x (208x82)

    const int b    = blockIdx.x;
    const int base = base0 + 2 * (int)blockIdx.y;
    float* out     = blockIdx.y ? out1 : out0;

    const int tid  = threadIdx.x;
    const int wave = tid >> 5;
    const int lane = tid & 31;
    const int hi   = lane >> 4;
    const int l15  = lane & 15;

    // ---- stage comb = [x_t | (R*)H] into LDS buffer 0, zero padded --------
    for (int i = tid; i < BUFSZ; i += 256) {
        int v = i / LSTRIDE, c = i - v * LSTRIDE;
        float val = 0.0f;
        if (v < N_NODES) {
            if (c < 2) {
                val = xt[(b * N_NODES + v) * 2 + c];
            } else if (c < CIN) {
                float h = H[((size_t)b * N_NODES + v) * HID + (c - 2)];
                if (Rg) h *= Rg[((size_t)b * N_NODES + v) * HID + (c - 2)];
                val = h;
            }
        }
        lds[i] = val;
    }
    __syncthreads();

    const v8f zero8 = {};
    v8f acc[2][4];                        // projection accumulators (VGPRs)
#pragma unroll
    for (int ti = 0; ti < 2; ++ti)
#pragma unroll
        for (int nt = 0; nt < 4; ++nt) acc[ti][nt] = zero8;

#pragma unroll 1
    for (int br = 0; br < 2; ++br) {
        const float* Ab = (br ? ATpad : Apad) + (size_t)b * NPAD * NPAD;
        const float* Pb = br ? pA1T : pA0T;

        // ---- two propagation passes: comb->h1, h1->h2 ---------------------
#pragma unroll 1
        for (int pass = 0; pass < 2; ++pass) {
            const int srcOff = pass * BUFSZ;
            const int dstOff = (pass + 1) * BUFSZ;
#pragma unroll 1
            for (int rt = wave; rt < 13; rt += 8) {
                const int m0 = rt * 16;
                const int m  = m0 + l15;
                const float* Ar = Ab + (size_t)m * NPAD;
                const float* Pr = Pb + (size_t)m * NPAD;
                __builtin_prefetch(Ar, 0, 1);
                __builtin_prefetch(Pr, 0, 1);
                v8f d[5];
#pragma unroll
                for (int nt = 0; nt < 5; ++nt) d[nt] = zero8;

#pragma unroll 1
                for (int k0 = 0; k0 < NPAD; k0 += 4) {
                    const int ka = k0 + 2 * hi;            // even
                    v2f a1 = *(const v2f*)(Ar + ka);       // global_load_b64
                    v2f a2 = *(const v2f*)(Pr + ka);
                    v2f bb[5];
                    const int s0 = srcOff + ka * LSTRIDE + l15;
#pragma unroll
                    for (int nt = 0; nt < 5; ++nt) {
                        bb[nt].x = lds[s0 + nt * 16];
                        bb[nt].y = lds[s0 + LSTRIDE + nt * 16];
                    }
#pragma unroll
                    for (int nt = 0; nt < 5; ++nt)
                        d[nt] = __builtin_amdgcn_wmma_f32_16x16x4_f32(
                            false, a1, false, bb[nt], (short)0, d[nt],
                            false, false);
#pragma unroll
                    for (int nt = 0; nt < 5; ++nt)
                        d[nt] = __builtin_amdgcn_wmma_f32_16x16x4_f32(
                            false, a2, false, bb[nt], (short)0, d[nt],
                            false, false);
                }
                // h_next = ALPHA*comb + D (padding stays zero: A rows 207 and
                // comb row 207 / cols>=66 are all zero)
#pragma unroll
                for (int nt = 0; nt < 5; ++nt)
#pragma unroll
                    for (int i = 0; i < 8; ++i) {
                        int row = m0 + i + 8 * hi;
                        int col = nt * 16 + l15;
                        lds[dstOff + row * LSTRIDE + col] =
                            d[nt][i] + ALPHA_C * lds[row * LSTRIDE + col];
                    }
            }
            __syncthreads();
        }

        // ---- projection: acc += [comb|h1|h2] @ Wpad[base+br] --------------
        const float* W = Wpad + (size_t)(base + br) * (3 * KPADP) * HID;
#pragma unroll
        for (int ti = 0; ti < 2; ++ti) {
            int rt = wave + ti * 8;
            if (rt < 13) {
                const int mrow = rt * 16 + l15;
#pragma unroll 1
                for (int seg = 0; seg < 3; ++seg) {
                    const int hOff = seg * BUFSZ + mrow * LSTRIDE;
                    const float* Wseg = W + seg * KPADP * HID;
#pragma unroll 1
                    for (int k0 = 0; k0 < KPADP; k0 += 4) {
                        const int ka = k0 + 2 * hi;        // even
                        v2f a = *(const v2f*)&lds[hOff + ka];  // ds_load_b64
#pragma unroll
                        for (int nt = 0; nt < 4; ++nt) {
                            v2f bb;
                            bb.x = Wseg[ka * HID + nt * 16 + l15];
                            bb.y = Wseg[(ka + 1) * HID + nt * 16 + l15];
                            acc[ti][nt] = __builtin_amdgcn_wmma_f32_16x16x4_f32(
                                false, a, false, bb, (short)0, acc[ti][nt],
                                false, false);
                        }
                    }
                }
            }
        }
        __syncthreads();   // before branch B overwrites h1/h2
    }

    // ---- bias + activation + store ---------------------------------------
    const float* bA = gB + (size_t)base * HID;
    const float* bB = gB + (size_t)(base + 1) * HID;
#pragma unroll
    for (int ti = 0; ti < 2; ++ti) {
        int rt = wave + ti * 8;
        if (rt < 13) {
            int m0 = rt * 16;
#pragma unroll
            for (int nt = 0; nt < 4; ++nt) {
                int col = nt * 16 + l15;
                float bias = bA[col] + bB[col];
#pragma unroll
                for (int i = 0; i < 8; ++i) {
                    int row = m0 + i + 8 * hi;
                    if (row < N_NODES) {
                        float v = acc[ti][nt][i] + bias;
                        v = act ? tanhf(v) : 1.0f / (1.0f + expf(-v));
                        out[((size_t)b * N_NODES + row) * HID + col] = v;
                    }
                }
            }
        }
    }
}

// ---------------------------------------------------------------------------
__global__ void update_h_kernel(float* __restrict__ H,
                                const float* __restrict__ Z,
                                const float* __restrict__ C, int n) {
    int i = blockIdx.x * blockDim.x + threadIdx.x;
    if (i < n) {
        float z = Z[i];
        H[i] = z * H[i] + (1.0f - z) * C[i];
    }
}

// ---------------------------------------------------------------------------
// Hd = gelu( interleave(gelu(H1*wL)+gelu(H2*wL), gelu(H1*wH)+gelu(H2*wH))
//            @ camW + camB );  one block per (b,v), 64 threads
// ---------------------------------------------------------------------------
__global__ void cam_kernel(const float* __restrict__ H1,
                           const float* __restrict__ H2,
                           const float* __restrict__ wL,
                           const float* __restrict__ wH,
                           const float* __restrict__ camW,   // (128,64)
                           const float* __restrict__ camB,   // (64)
                           float* __restrict__ out) {
    __shared__ float cv[2 * HID];
    int bv = blockIdx.x;
    int v  = bv % N_NODES;
    int c  = threadIdx.x;
    float h1 = H1[(size_t)bv * HID + c];
    float h2 = H2[(size_t)bv * HID + c];
    float l  = geluf(h1 * wL[v * HID + c]) + geluf(h2 * wL[v * HID + c]);
    float h  = geluf(h1 * wH[v * HID + c]) + geluf(h2 * wH[v * HID + c]);
    cv[2 * c]     = l;
    cv[2 * c + 1] = h;
    __syncthreads();
    float s = camB[c];
    for (int k = 0; k < 2 * HID; ++k) s += cv[k] * camW[k * HID + c];
    out[(size_t)bv * HID + c] = geluf(s);
}

// ---------------------------------------------------------------------------
__global__ void fc_kernel(const float* __restrict__ H,
                          const float* __restrict__ fcW,
                          const float* __restrict__ fcb,
                          float* __restrict__ out, int t) {
    int idx = blockIdx.x * blockDim.x + threadIdx.x;
    if (idx >= BATCH * N_NODES) return;
    int b = idx / N_NODES, v = idx - b * N_NODES;
    const float* h = H + (size_t)idx * HID;
    float s = fcb[0];
#pragma unroll
    for (int c = 0; c < HID; ++c) s += h[c] * fcW[c];
    out[((size_t)b * 12 + t) * N_NODES + v] = s;
}

// ---------------------------------------------------------------------------
extern "C" void kernel_launch(void* const* d_in, const int* in_sizes, int n_in,
                              void* d_out, int out_size, void* d_ws,
                              size_t ws_size, hipStream_t stream) {
    (void)in_sizes; (void)n_in; (void)out_size; (void)ws_size;
    const float* input = (const float*)d_in[0];
    const float* md1   = (const float*)d_in[1];
    const float* md2   = (const float*)d_in[2];
    const float* ycl   = (const float*)d_in[3];
    const float* pA0   = (const float*)d_in[4];
    const float* pA1   = (const float*)d_in[5];
    const float* gW    = (const float*)d_in[6];
    const float* gB    = (const float*)d_in[7];
    const float* wL    = (const float*)d_in[8];
    const float* wH    = (const float*)d_in[9];
    const float* camHW = (const float*)d_in[10];
    const float* camHb = (const float*)d_in[11];
    // d_in[12]/[13] = camC_W/camC_b: dead code in the reference (Cd unused)
    const float* fcW   = (const float*)d_in[14];
    const float* fcb   = (const float*)d_in[15];
    float* out = (float*)d_out;

    float* ws = (float*)d_ws;
    size_t o = 0;
    const size_t BN   = (size_t)BATCH * N_NODES;
    const size_t BNH  = BN * HID;
    const size_t APAD = (size_t)BATCH * NPAD * NPAD;
    float* rs1  = ws + o; o += BN;
    float* cs1  = ws + o; o += BN;
    float* rs2  = ws + o; o += BN;
    float* cs2  = ws + o; o += BN;
    float* XT   = ws + o; o += BN * 2;
    float* Z    = ws + o; o += BNH;
    float* Rg   = ws + o; o += BNH;
    float* H1   = ws + o; o += BNH;
    float* H2   = ws + o; o += BNH;
    float* HD   = ws + o; o += BNH;
    float* C1   = ws + o; o += BNH;
    float* C2   = ws + o; o += BNH;
    float* CD   = ws + o; o += BNH;
    float* pA0T = ws + o; o += NPAD * NPAD;
    float* pA1T = ws + o; o += NPAD * NPAD;
    float* Wpad = ws + o; o += (size_t)18 * 3 * KPADP * HID;
    float* Apad = ws + o; o += APAD;     // recycled: encoder md1, decoder md2
    float* ATpad = ws + o; o += APAD;

    const size_t ldsB = (size_t)3 * BUFSZ * sizeof(float);
    const int gBN   = (int)((BN + 255) / 256);
    const int gBNH  = (int)((BNH + 255) / 256);
    const int gPA   = (NPAD * NPAD + 255) / 256;
    const int gWp   = (18 * 3 * KPADP * HID + 255) / 256;

    sums_kernel<<<dim3(BATCH, 2), 224, 0, stream>>>(md1, rs1, cs1);
    sums_kernel<<<dim3(BATCH, 2), 224, 0, stream>>>(md2, rs2, cs2);
    prep_pa_kernel<<<gPA, 256, 0, stream>>>(pA0, pA0T);
    prep_pa_kernel<<<gPA, 256, 0, stream>>>(pA1, pA1T);
    prep_w_kernel<<<gWp, 256, 0, stream>>>(gW, Wpad);
    fill0_kernel<<<gBNH, 256, 0, stream>>>(H1, (int)BNH);
    fill0_kernel<<<gBNH, 256, 0, stream>>>(H2, (int)BNH);

    // ---- encoder phase: panels from md1 -----------------------------------
    prep_adj_kernel<<<dim3(BATCH, NPAD), 256, 0, stream>>>(md1, rs1, cs1,
                                                           Apad, ATpad);
    for (int enc = 0; enc < 2; ++enc) {
        float* Hc = enc ? H2 : H1;
        float* Cc = enc ? C2 : C1;
        float sgn = enc ? 1.0f : -1.0f;   // xAD : xD
        int base  = enc * 6;
        for (int t = 0; t < 6; ++t) {
            build_xt_kernel<<<gBN, 256, 0, stream>>>(input, nullptr, nullptr,
                                                     XT, t, 0, sgn);
            gate_kernel<<<dim3(BATCH, 2), 256, ldsB, stream>>>(
                XT, Hc, nullptr, Apad, ATpad, pA0T, pA1T, Wpad, gB,
                base, 0, Z, Rg);                       // z, r
            gate_kernel<<<dim3(BATCH, 1), 256, ldsB, stream>>>(
                XT, Hc, Rg, Apad, ATpad, pA0T, pA1T, Wpad, gB,
                base + 4, 1, Cc, Cc);                  // C (tanh)
            update_h_kernel<<<gBNH, 256, 0, stream>>>(Hc, Z, Cc, (int)BNH);
        }
    }

    // ---- IDWT combine + camH -> decoder initial hidden state --------------
    cam_kernel<<<(int)BN, HID, 0, stream>>>(H1, H2, wL, wH, camHW, camHb, HD);

    // ---- decoder phase: panels from md2 -----------------------------------
    prep_adj_kernel<<<dim3(BATCH, NPAD), 256, 0, stream>>>(md2, rs2, cs2,
                                                           Apad, ATpad);
    for (int t = 0; t < 12; ++t) {
        build_xt_kernel<<<gBN, 256, 0, stream>>>(nullptr, ycl, out, XT, t, 1,
                                                 0.0f);
        gate_kernel<<<dim3(BATCH, 2), 256, ldsB, stream>>>(
            XT, HD, nullptr, Apad, ATpad, pA0T, pA1T, Wpad, gB, 12, 0, Z, Rg);
        gate_kernel<<<dim3(BATCH, 1), 256, ldsB, stream>>>(
            XT, HD, Rg, Apad, ATpad, pA0T, pA1T, Wpad, gB, 16, 1, CD, CD);
        update_h_kernel<<<gBNH, 256, 0, stream>>>(HD, Z, CD, (int)BNH);
        fc_kernel<<<gBN, 256, 0, stream>>>(HD, fcW, fcb, out, t);
    }
}